// BatchSamplingLoss_31104153158231
// MI455X (gfx1250) — compile-verified
//
#include <hip/hip_runtime.h>
#include <math.h>

#define B_BATCH 32
#define IMG_H   1024
#define IMG_W   2048
#define TPB     256
#define PI_F    3.14159265358979323846f

// ---- CDNA5 async global->LDS path (ASYNCcnt-tracked data mover) ----------
#if defined(__HIP_DEVICE_COMPILE__) && __has_builtin(__builtin_amdgcn_global_load_async_to_lds_b32)
#define USE_ASYNC 1
#else
#define USE_ASYNC 0
#endif

#if defined(__HIP_DEVICE_COMPILE__)
typedef __attribute__((address_space(1))) int as1_int;   // '__device__ int *'
typedef __attribute__((address_space(3))) int as3_int;   // '__shared__ int *'
#endif

__device__ __forceinline__ void async_copy_dword(const float* g, float* l) {
#if USE_ASYNC
  __builtin_amdgcn_global_load_async_to_lds_b32((as1_int*)g, (as3_int*)l, 0, 0);
#else
  (void)g; (void)l;
#endif
}

__device__ __forceinline__ void wait_async0() {
#if USE_ASYNC
#if __has_builtin(__builtin_amdgcn_s_wait_asynccnt)
  __builtin_amdgcn_s_wait_asynccnt(0);
#else
  asm volatile("s_wait_asynccnt 0" ::: "memory");
#endif
  asm volatile("" ::: "memory");
#endif
}

// ---- Main kernel (defined first so the disasm snippet shows its prologue):
// transform + equirect project + bilinear sample + masked L2 loss ---------
__global__ void __launch_bounds__(TPB)
sample_loss_k(const float* __restrict__ xyz, const float* __restrict__ rgb,
              const float* __restrict__ img, const float* __restrict__ M,
              float* __restrict__ gacc, int N) {
  __shared__ float sacc[2 * B_BATCH];
#if USE_ASYNC
  __shared__ float stage[2][TPB * 6];  // double-buffered per-thread staging (xyz+rgb)
#endif
  const int tid = threadIdx.x;
  if (tid < 2 * B_BATCH) sacc[tid] = 0.0f;
  __syncthreads();

  const int gtid   = blockIdx.x * TPB + tid;
  const int stride = gridDim.x * TPB;
  const int iters  = (N + stride - 1) / stride;  // uniform trip count: all lanes stay active

#if USE_ASYNC
  {  // prologue: async-prefetch first point into buffer 0 (clamped address)
    int idx = gtid;
    int c   = (idx < N) ? idx : 0;
    const float* gx = xyz + (size_t)c * 3;
    const float* gr = rgb + (size_t)c * 3;
    float* l = &stage[0][tid * 6];
    async_copy_dword(gx + 0, l + 0); async_copy_dword(gx + 1, l + 1); async_copy_dword(gx + 2, l + 2);
    async_copy_dword(gr + 0, l + 3); async_copy_dword(gr + 1, l + 4); async_copy_dword(gr + 2, l + 5);
  }
#endif

  for (int it = 0; it < iters; ++it) {
    const int  idx = gtid + it * stride;
    const bool inb = idx < N;
    float px, py, pz, pr_, pg_, pb_;
#if USE_ASYNC
    wait_async0();  // my wave's loads for stage[it&1] done (self-staged: no barrier needed)
    {
      const float* l = &stage[it & 1][tid * 6];
      px = l[0]; py = l[1]; pz = l[2];
      pr_ = l[3]; pg_ = l[4]; pb_ = l[5];
    }
    if (it + 1 < iters) {  // prefetch next point into the other buffer; overlaps compute below
      int nidx = gtid + (it + 1) * stride;
      int c    = (nidx < N) ? nidx : 0;
      const float* gx = xyz + (size_t)c * 3;
      const float* gr = rgb + (size_t)c * 3;
      float* l = &stage[(it + 1) & 1][tid * 6];
      async_copy_dword(gx + 0, l + 0); async_copy_dword(gx + 1, l + 1); async_copy_dword(gx + 2, l + 2);
      async_copy_dword(gr + 0, l + 3); async_copy_dword(gr + 1, l + 4); async_copy_dword(gr + 2, l + 5);
    }
#else
    {
      int c = inb ? idx : 0;
      px = xyz[(size_t)c * 3 + 0]; py = xyz[(size_t)c * 3 + 1]; pz = xyz[(size_t)c * 3 + 2];
      pr_ = rgb[(size_t)c * 3 + 0]; pg_ = rgb[(size_t)c * 3 + 1]; pb_ = rgb[(size_t)c * 3 + 2];
    }
#endif

#pragma unroll 4
    for (int b = 0; b < B_BATCH; ++b) {
      float l = 0.0f, cnt = 0.0f;
      if (inb) {
        const float* m = M + b * 12;  // uniform address -> scalar loads
        float nx = fmaf(m[0], px, fmaf(m[1], py, fmaf(m[2], pz, m[9])));
        float ny = fmaf(m[3], px, fmaf(m[4], py, fmaf(m[5], pz, m[10])));
        float nz = fmaf(m[6], px, fmaf(m[7], py, fmaf(m[8], pz, m[11])));
        float theta = atan2f(sqrtf(fmaf(nx, nx, ny * ny)), nz);   // [0, pi]
        float phi   = atan2f(ny, nx) + PI_F;                      // [0, 2pi]
        // pixel coords (align_corners=False):
        float xf = (float)IMG_W - phi * ((float)IMG_W / (2.0f * PI_F)) - 0.5f;
        float yf = theta * ((float)IMG_H / PI_F) - 0.5f;
        float x0f = floorf(xf), y0f = floorf(yf);
        float wx = xf - x0f, wy = yf - y0f;
        int ix0 = (int)x0f, iy0 = (int)y0f;
        int ix1 = ix0 + 1,  iy1 = iy0 + 1;
        bool vx0 = (ix0 >= 0) && (ix0 < IMG_W);
        bool vx1 = (ix1 >= 0) && (ix1 < IMG_W);
        bool vy0 = (iy0 >= 0) && (iy0 < IMG_H);
        bool vy1 = (iy1 >= 0) && (iy1 < IMG_H);
        int cx0 = min(max(ix0, 0), IMG_W - 1), cx1 = min(max(ix1, 0), IMG_W - 1);
        int cy0 = min(max(iy0, 0), IMG_H - 1), cy1 = min(max(iy1, 0), IMG_H - 1);
        float w00 = ((vx0 & vy0) ? 1.0f : 0.0f) * (1.0f - wx) * (1.0f - wy);
        float w10 = ((vx1 & vy0) ? 1.0f : 0.0f) * wx * (1.0f - wy);
        float w01 = ((vx0 & vy1) ? 1.0f : 0.0f) * (1.0f - wx) * wy;
        float w11 = ((vx1 & vy1) ? 1.0f : 0.0f) * wx * wy;
        const float* p00 = img + ((size_t)cy0 * IMG_W + cx0) * 3;  // L2-resident gathers
        const float* p10 = img + ((size_t)cy0 * IMG_W + cx1) * 3;
        const float* p01 = img + ((size_t)cy1 * IMG_W + cx0) * 3;
        const float* p11 = img + ((size_t)cy1 * IMG_W + cx1) * 3;
        float sr = w00 * p00[0] + w10 * p10[0] + w01 * p01[0] + w11 * p11[0];
        float sg = w00 * p00[1] + w10 * p10[1] + w01 * p01[1] + w11 * p11[1];
        float sb = w00 * p00[2] + w10 * p10[2] + w01 * p01[2] + w11 * p11[2];
        bool valid = !((sr == 0.0f) & (sg == 0.0f) & (sb == 0.0f));
        float dr = sr - pr_, dg = sg - pg_, db = sb - pb_;
        float vf = valid ? 1.0f : 0.0f;
        l   = sqrtf(fmaf(dr, dr, fmaf(dg, dg, db * db))) * vf;
        cnt = vf;
      }
      // wave32 butterfly reduction, then single-lane LDS atomic
      for (int off = 16; off > 0; off >>= 1) {
        l   += __shfl_xor(l,   off, 32);
        cnt += __shfl_xor(cnt, off, 32);
      }
      if ((tid & 31) == 0) {
        atomicAdd(&sacc[b], l);                 // ds_add_f32
        atomicAdd(&sacc[B_BATCH + b], cnt);
      }
    }
  }
  __syncthreads();
  if (tid < 2 * B_BATCH) atomicAdd(&gacc[tid], sacc[tid]);  // global_atomic_add_f32
}

// ---- Kernel 1: per-pose transform precompute + accumulator zeroing -------
// M[b*12 + 3r+c] = Rm[r][c];  M[b*12 + 9 + r] = -(Rm[r] . t[b])
__global__ void precompute_k(const float* __restrict__ trans,
                             const float* __restrict__ yaw,
                             const float* __restrict__ pitch,
                             const float* __restrict__ roll,
                             float* __restrict__ M, float* __restrict__ gacc) {
  int t = threadIdx.x;
  if (t < 2 * B_BATCH) gacc[t] = 0.0f;  // harness does not re-zero ws between replays
  if (t < B_BATCH) {
    float cy = cosf(yaw[t]),   sy = sinf(yaw[t]);
    float cp = cosf(pitch[t]), sp = sinf(pitch[t]);
    float cr = cosf(roll[t]),  sr = sinf(roll[t]);
    // Rm = RZ(yaw) @ RY(pitch) @ RX(roll)  (closed form)
    float r00 = cy * cp;
    float r01 = cy * sp * sr - sy * cr;
    float r02 = cy * sp * cr + sy * sr;
    float r10 = sy * cp;
    float r11 = sy * sp * sr + cy * cr;
    float r12 = sy * sp * cr - cy * sr;
    float r20 = -sp;
    float r21 = cp * sr;
    float r22 = cp * cr;
    float tx = trans[t * 3 + 0], ty = trans[t * 3 + 1], tz = trans[t * 3 + 2];
    float* m = M + t * 12;
    m[0] = r00; m[1] = r01; m[2] = r02;
    m[3] = r10; m[4] = r11; m[5] = r12;
    m[6] = r20; m[7] = r21; m[8] = r22;
    m[9]  = -(r00 * tx + r01 * ty + r02 * tz);
    m[10] = -(r10 * tx + r11 * ty + r12 * tz);
    m[11] = -(r20 * tx + r21 * ty + r22 * tz);
  }
}

// ---- Kernel 3: finalize -------------------------------------------------
__global__ void finalize_k(const float* __restrict__ gacc, float* __restrict__ out) {
  int b = threadIdx.x;
  float v = 0.0f;
  if (b < B_BATCH) {
    v = gacc[b] / gacc[B_BATCH + b];
    out[1 + b] = v;
  }
  for (int off = 16; off > 0; off >>= 1) v += __shfl_xor(v, off, 32);
  if (b == 0) out[0] = v;
}

extern "C" void kernel_launch(void* const* d_in, const int* in_sizes, int n_in,
                              void* d_out, int out_size, void* d_ws, size_t ws_size,
                              hipStream_t stream) {
  const float* xyz   = (const float*)d_in[0];
  const float* rgb   = (const float*)d_in[1];
  const float* img   = (const float*)d_in[2];
  const float* trans = (const float*)d_in[3];
  const float* yaw   = (const float*)d_in[4];
  const float* pitch = (const float*)d_in[5];
  const float* roll  = (const float*)d_in[6];
  const int N = in_sizes[0] / 3;

  float* wsf  = (float*)d_ws;
  float* M    = wsf;         // 32*12 floats
  float* gacc = wsf + 384;   // 64 floats: [0..31]=loss sums, [32..63]=counts
  float* out  = (float*)d_out;

  // size grid for exactly 2 grid-stride iterations per thread:
  // ~99.9% lane utilization + meaningful double-buffered async prefetch,
  // while still launching ~3K wave32s for gather-latency hiding.
  int blocks = (N + 2 * TPB - 1) / (2 * TPB);
  if (blocks < 1) blocks = 1;

  hipLaunchKernelGGL(precompute_k, dim3(1), dim3(64), 0, stream, trans, yaw, pitch, roll, M, gacc);
  hipLaunchKernelGGL(sample_loss_k, dim3(blocks), dim3(TPB), 0, stream, xyz, rgb, img, M, gacc, N);
  hipLaunchKernelGGL(finalize_k, dim3(1), dim3(32), 0, stream, gacc, out);
}